// RelativePositionAttention_55018531062341
// MI455X (gfx1250) — compile-verified
//
#include <hip/hip_runtime.h>

typedef __attribute__((ext_vector_type(2))) float v2f;
typedef __attribute__((ext_vector_type(8))) float v8f;

#define WMMA_F32(a, b, c) \
  __builtin_amdgcn_wmma_f32_16x16x4_f32(false, (a), false, (b), (short)0, (c), false, false)

constexpr int BATCH = 4;
constexpr int SEQ   = 1024;
constexpr int EMB   = 1024;
constexpr int NH    = 16;
constexpr int HD    = 64;
constexpr int NREL  = 65;           // 2*32+1
constexpr int MROWS = BATCH * SEQ;  // 4096

// ---------------------------------------------------------------------------
// fp32 WMMA GEMM:  C[M,N] = A[M,K] @ W[K,N] + bias
// Block tile 128(M) x 128(N), BK = 32. 8 waves arranged 4(M) x 2(N),
// each wave owns 32 x 64 (2 x 4 fragments). Global tiles loaded as b128;
// B tile stored transposed in LDS; strides multiple-of-4 floats so A-tile
// stores are single ds_store_b128 and frag loads are aligned ds_load_b64.
// qkv != 0: scatter output into [B, H, S, D] head layout.
// ---------------------------------------------------------------------------
__global__ __launch_bounds__(256)
void proj_gemm(const float* __restrict__ A, const float* __restrict__ W,
               const float* __restrict__ bias, float* __restrict__ Cout,
               int N, int K, int qkv) {
  const int tid  = threadIdx.x;
  const int lane = tid & 31, wid = tid >> 5;
  const int l15  = lane & 15, h = lane >> 4;
  const int wm   = wid >> 1, wn = wid & 1;          // 4 x 2 wave grid
  const int m0   = blockIdx.y * 128, n0 = blockIdx.x * 128;

  __shared__ __align__(16) float As[128][36];       // As[m][k]
  __shared__ __align__(16) float Wt[128][36];       // Wt[n][k]  (transposed)

  v8f acc[2][4] = {};

  for (int k0 = 0; k0 < K; k0 += 32) {
#pragma unroll
    for (int it = 0; it < 4; ++it) {                // A tile 128x32, b128 loads
      int idx = tid + it * 256;                     // 1024 float4
      int r = idx >> 3, c4 = idx & 7;
      float4 v = *(const float4*)&A[(size_t)(m0 + r) * K + k0 + c4 * 4];
      *(float4*)&As[r][c4 * 4] = v;
    }
#pragma unroll
    for (int it = 0; it < 4; ++it) {                // W tile 32x128 -> Wt
      int idx = tid + it * 256;
      int r = idx >> 5, c4 = idx & 31;
      float4 v = *(const float4*)&W[(size_t)(k0 + r) * N + n0 + c4 * 4];
      Wt[c4 * 4 + 0][r] = v.x;
      Wt[c4 * 4 + 1][r] = v.y;
      Wt[c4 * 4 + 2][r] = v.z;
      Wt[c4 * 4 + 3][r] = v.w;
    }
    if (k0 + 32 < K) {                              // prefetch next K tile
      __builtin_prefetch(&A[(size_t)(m0 + (tid & 127)) * K + k0 + 32], 0, 0);
      __builtin_prefetch(&W[(size_t)(k0 + 32 + (tid & 31)) * N + n0], 0, 0);
    }
    __syncthreads();

#pragma unroll
    for (int kk = 0; kk < 32; kk += 4) {
      v2f a[2], b[4];
#pragma unroll
      for (int i = 0; i < 2; ++i)
        a[i] = *(const v2f*)&As[wm * 32 + i * 16 + l15][kk + 2 * h];
#pragma unroll
      for (int j = 0; j < 4; ++j)
        b[j] = *(const v2f*)&Wt[wn * 64 + j * 16 + l15][kk + 2 * h];
#pragma unroll
      for (int i = 0; i < 2; ++i)
#pragma unroll
        for (int j = 0; j < 4; ++j)
          acc[i][j] = WMMA_F32(a[i], b[j], acc[i][j]);
    }
    __syncthreads();
  }

#pragma unroll
  for (int i = 0; i < 2; ++i) {
#pragma unroll
    for (int j = 0; j < 4; ++j) {
      int n = n0 + wn * 64 + j * 16 + l15;
      float bv = bias[n];
#pragma unroll
      for (int e = 0; e < 8; ++e) {
        int m = m0 + wm * 32 + i * 16 + h * 8 + e;
        float v = acc[i][j][e] + bv;
        if (qkv) {
          int b_ = m >> 10, s = m & 1023;           // m = b*S + s
          int hh = n >> 6, d = n & 63;              // n = h*D + d
          Cout[(((size_t)(b_ * NH + hh) * SEQ) + s) * HD + d] = v;
        } else {
          Cout[(size_t)m * N + n] = v;
        }
      }
    }
  }
}

// ---------------------------------------------------------------------------
// Qrel[row, r] = Q[row, :] . rel_k[r, :]   (row = bh*S + s, r = 0..64)
// ---------------------------------------------------------------------------
__global__ __launch_bounds__(128)
void qrel_kernel(const float* __restrict__ Q, const float* __restrict__ relk,
                 float* __restrict__ Qrel) {
  const int row = blockIdx.x;
  __shared__ float q[HD];
  if (threadIdx.x < HD) q[threadIdx.x] = Q[(size_t)row * HD + threadIdx.x];
  __syncthreads();
  int r = threadIdx.x;
  if (r < NREL) {
    float acc = 0.f;
#pragma unroll
    for (int d = 0; d < HD; ++d) acc += q[d] * relk[r * HD + d];
    Qrel[(size_t)row * NREL + r] = acc;
  }
}

// ---------------------------------------------------------------------------
// scores[bh, s, t] = (Q.K^T)/8 + Qrel[bh*S+s, clip(t-s,-32,32)+32]
// Block tile 64(s) x 128(t); 8 waves 2(M) x 4(N), each 32x32 (2x2 frags).
// Full D=64 held in LDS; K rows stored row-major => B frags contiguous.
// ---------------------------------------------------------------------------
__global__ __launch_bounds__(256)
void scores_kernel(const float* __restrict__ Q, const float* __restrict__ Kmat,
                   const float* __restrict__ Qrel, float* __restrict__ attn) {
  const int bh = blockIdx.z;
  const int sT = blockIdx.y, tT = blockIdx.x;
  const int tid = threadIdx.x, lane = tid & 31, wid = tid >> 5;
  const int l15 = lane & 15, h = lane >> 4;
  const int wm = wid >> 2, wn = wid & 3;            // 2 x 4 wave grid

  const float* Qb = Q    + (size_t)bh * SEQ * HD;
  const float* Kb = Kmat + (size_t)bh * SEQ * HD;

  __shared__ __align__(16) float Qs[64][68];        // Qs[s][d]
  __shared__ __align__(16) float Ks[128][68];       // Ks[t][d]

#pragma unroll
  for (int it = 0; it < 4; ++it) {                  // Q tile 64x64, b128
    int idx = tid + it * 256;                       // 1024 float4
    int r = idx >> 4, c4 = idx & 15;
    float4 v = *(const float4*)&Qb[(size_t)(sT * 64 + r) * HD + c4 * 4];
    *(float4*)&Qs[r][c4 * 4] = v;
  }
#pragma unroll
  for (int it = 0; it < 8; ++it) {                  // K tile 128x64, b128
    int idx = tid + it * 256;                       // 2048 float4
    int r = idx >> 4, c4 = idx & 15;
    float4 v = *(const float4*)&Kb[(size_t)(tT * 128 + r) * HD + c4 * 4];
    *(float4*)&Ks[r][c4 * 4] = v;
  }
  __syncthreads();

  v8f acc[2][2] = {};
#pragma unroll
  for (int kk = 0; kk < HD; kk += 4) {
    v2f a[2], b[2];
#pragma unroll
    for (int i = 0; i < 2; ++i)
      a[i] = *(const v2f*)&Qs[wm * 32 + i * 16 + l15][kk + 2 * h];
#pragma unroll
    for (int j = 0; j < 2; ++j)                     // B[k=d][n=t] = K[t][d]
      b[j] = *(const v2f*)&Ks[wn * 32 + j * 16 + l15][kk + 2 * h];
#pragma unroll
    for (int i = 0; i < 2; ++i)
#pragma unroll
      for (int j = 0; j < 2; ++j)
        acc[i][j] = WMMA_F32(a[i], b[j], acc[i][j]);
  }

  const float scale = 0.125f;                       // 1/sqrt(64)
#pragma unroll
  for (int i = 0; i < 2; ++i) {
#pragma unroll
    for (int j = 0; j < 2; ++j) {
      int t = tT * 128 + wn * 32 + j * 16 + l15;
#pragma unroll
      for (int e = 0; e < 8; ++e) {
        int s = sT * 64 + wm * 32 + i * 16 + h * 8 + e;
        int rel = t - s;
        rel = rel < -32 ? -32 : (rel > 32 ? 32 : rel);
        float v = acc[i][j][e] * scale +
                  Qrel[((size_t)bh * SEQ + s) * NREL + rel + 32];
        attn[(((size_t)bh * SEQ) + s) * SEQ + t] = v;
      }
    }
  }
}

// ---------------------------------------------------------------------------
// In-place row softmax over 1024 elements; one 256-thread block per row.
// b128 row I/O (4 contiguous floats per thread).
// ---------------------------------------------------------------------------
__global__ __launch_bounds__(256)
void softmax_kernel(float* __restrict__ attn) {
  float* p = attn + (size_t)blockIdx.x * SEQ;
  const int tid = threadIdx.x;
  __shared__ float red[256];

  float4 x = *(const float4*)&p[tid * 4];
  float mx = fmaxf(fmaxf(x.x, x.y), fmaxf(x.z, x.w));
  red[tid] = mx; __syncthreads();
  for (int s = 128; s > 0; s >>= 1) {
    if (tid < s) red[tid] = fmaxf(red[tid], red[tid + s]);
    __syncthreads();
  }
  mx = red[0]; __syncthreads();

  x.x = __expf(x.x - mx); x.y = __expf(x.y - mx);
  x.z = __expf(x.z - mx); x.w = __expf(x.w - mx);
  float sum = (x.x + x.y) + (x.z + x.w);
  red[tid] = sum; __syncthreads();
  for (int s = 128; s > 0; s >>= 1) {
    if (tid < s) red[tid] += red[tid + s];
    __syncthreads();
  }
  float inv = 1.f / red[0];
  x.x *= inv; x.y *= inv; x.z *= inv; x.w *= inv;
  *(float4*)&p[tid * 4] = x;
}

// ---------------------------------------------------------------------------
// ctx[bh] = attn[bh] (SxS) @ V[bh] (SxD); output relayout to [B, S, E].
// Block tile 256(M) x 64(N), BK = 32. 8 waves stacked in M, each 32x64
// (2x4 frags). V tile transposed in LDS for contiguous B fragments.
// ---------------------------------------------------------------------------
__global__ __launch_bounds__(256)
void ctx_kernel(const float* __restrict__ attn, const float* __restrict__ V,
                float* __restrict__ ctx) {
  const int bh = blockIdx.z;
  const int mT = blockIdx.x;
  const int tid = threadIdx.x, lane = tid & 31, wid = tid >> 5;
  const int l15 = lane & 15, h = lane >> 4;
  const int wm = wid;                               // 8 waves in M

  const float* Ab = attn + (size_t)bh * SEQ * SEQ;
  const float* Vb = V    + (size_t)bh * SEQ * HD;

  __shared__ __align__(16) float As[256][36];       // As[s][k]
  __shared__ __align__(16) float Vt[64][36];        // Vt[d][k] (transposed)

  v8f acc[2][4] = {};

  for (int k0 = 0; k0 < SEQ; k0 += 32) {
#pragma unroll
    for (int it = 0; it < 8; ++it) {                // attn tile 256x32, b128
      int idx = tid + it * 256;                     // 2048 float4
      int r = idx >> 3, c4 = idx & 7;
      float4 v = *(const float4*)&Ab[(size_t)(mT * 256 + r) * SEQ + k0 + c4 * 4];
      *(float4*)&As[r][c4 * 4] = v;
    }
#pragma unroll
    for (int it = 0; it < 2; ++it) {                // V tile 32x64 -> Vt
      int idx = tid + it * 256;                     // 512 float4
      int r = idx >> 4, c4 = idx & 15;
      float4 v = *(const float4*)&Vb[(size_t)(k0 + r) * HD + c4 * 4];
      Vt[c4 * 4 + 0][r] = v.x;
      Vt[c4 * 4 + 1][r] = v.y;
      Vt[c4 * 4 + 2][r] = v.z;
      Vt[c4 * 4 + 3][r] = v.w;
    }
    if (k0 + 32 < SEQ) {
      __builtin_prefetch(&Ab[(size_t)(mT * 256 + tid) * SEQ + k0 + 32], 0, 0);
      __builtin_prefetch(&Vb[(size_t)(k0 + 32 + (tid & 31)) * HD], 0, 0);
    }
    __syncthreads();

#pragma unroll
    for (int kk = 0; kk < 32; kk += 4) {
      v2f a[2], b[4];
#pragma unroll
      for (int i = 0; i < 2; ++i)
        a[i] = *(const v2f*)&As[wm * 32 + i * 16 + l15][kk + 2 * h];
#pragma unroll
      for (int j = 0; j < 4; ++j)
        b[j] = *(const v2f*)&Vt[j * 16 + l15][kk + 2 * h];
#pragma unroll
      for (int i = 0; i < 2; ++i)
#pragma unroll
        for (int j = 0; j < 4; ++j)
          acc[i][j] = WMMA_F32(a[i], b[j], acc[i][j]);
    }
    __syncthreads();
  }

  const int b_ = bh >> 4, hh = bh & 15;
#pragma unroll
  for (int i = 0; i < 2; ++i)
#pragma unroll
    for (int j = 0; j < 4; ++j) {
      int d = j * 16 + l15;
#pragma unroll
      for (int e = 0; e < 8; ++e) {
        int s = mT * 256 + wm * 32 + i * 16 + h * 8 + e;
        ctx[((size_t)(b_ * SEQ + s) * EMB) + hh * HD + d] = acc[i][j][e];
      }
    }
}

// ---------------------------------------------------------------------------
extern "C" void kernel_launch(void* const* d_in, const int* in_sizes, int n_in,
                              void* d_out, int out_size, void* d_ws, size_t ws_size,
                              hipStream_t stream) {
  const float* x    = (const float*)d_in[0];
  const float* Wq   = (const float*)d_in[1];
  const float* bq   = (const float*)d_in[2];
  const float* Wk   = (const float*)d_in[3];
  const float* bk   = (const float*)d_in[4];
  const float* Wv   = (const float*)d_in[5];
  const float* bv   = (const float*)d_in[6];
  const float* Wo   = (const float*)d_in[7];
  const float* bo   = (const float*)d_in[8];
  const float* relk = (const float*)d_in[9];

  float* out  = (float*)d_out;                          // [B,S,E]
  float* attn = out + (size_t)BATCH * SEQ * EMB;        // [B,H,S,S]

  const size_t qkvN = (size_t)BATCH * NH * SEQ * HD;    // 4.19M floats each
  float* ws   = (float*)d_ws;
  float* Q    = ws;
  float* K    = Q + qkvN;
  float* V    = K + qkvN;
  float* Qrel = V + qkvN;                               // [B*H*S, 65]
  float* ctx  = Qrel + (size_t)BATCH * NH * SEQ * NREL; // [B,S,E]

  dim3 blk(256);
  dim3 gproj(EMB / 128, MROWS / 128);                   // (8, 32)

  proj_gemm<<<gproj, blk, 0, stream>>>(x, Wq, bq, Q, EMB, EMB, 1);
  proj_gemm<<<gproj, blk, 0, stream>>>(x, Wk, bk, K, EMB, EMB, 1);
  proj_gemm<<<gproj, blk, 0, stream>>>(x, Wv, bv, V, EMB, EMB, 1);

  qrel_kernel<<<dim3(BATCH * NH * SEQ), dim3(128), 0, stream>>>(Q, relk, Qrel);

  scores_kernel<<<dim3(SEQ / 128, SEQ / 64, BATCH * NH), blk, 0, stream>>>(Q, K, Qrel, attn);

  softmax_kernel<<<dim3(BATCH * NH * SEQ), blk, 0, stream>>>(attn);

  ctx_kernel<<<dim3(SEQ / 256, 1, BATCH * NH), blk, 0, stream>>>(attn, V, ctx);

  proj_gemm<<<gproj, blk, 0, stream>>>(ctx, Wo, bo, out, EMB, EMB, 0);
}